// CrossModalCenterLoss_52166672778208
// MI455X (gfx1250) — compile-verified
//
#include <hip/hip_runtime.h>

// CrossModalCenterLoss, MI455X (gfx1250).
//
// loss = (1/B) * sum_b clip(||x_b - centers[label_b]||^2, 1e-12, 1e12) + (C-1)*1e-12
//
// Strategy: gather + fused squared distance. Difference fragments are fed
// through V_WMMA_F32_16X16X4_F32 (full f32 precision) as D = f * f^T + C;
// diag(D) after K-accumulation = the 16 per-row squared distances.
// For the symmetric product the A and B per-lane fragments are identical
// under the ISA-documented 16x4 f32 A layout, so no cross-lane staging.

typedef __attribute__((ext_vector_type(2))) float v2f;
typedef __attribute__((ext_vector_type(8))) float v8f;

#define FEAT_DIM     256
#define BATCH        4096
#define NUM_CLASSES  10000
#define CLAMP_MIN_F  1e-12f
#define CLAMP_MAX_F  1e12f

#define ROWS_PER_WAVE 16
#define NBLOCKS (BATCH / ROWS_PER_WAVE)   // 256 single-wave workgroups

__global__ __launch_bounds__(32) void cmcl_partial_kernel(
    const float* __restrict__ x,        // [BATCH, FEAT_DIM]
    const int*   __restrict__ labels,   // [BATCH]
    const float* __restrict__ centers,  // [NUM_CLASSES, FEAT_DIM]
    float*       __restrict__ partials) // [NBLOCKS]
{
    const int lane = threadIdx.x & 31;
    const int r    = lane & 15;         // matrix row handled by this lane
    const int half = lane >> 4;         // 0: K{0,1} of each chunk, 1: K{2,3}
    const int row  = blockIdx.x * ROWS_PER_WAVE + r;

    const int lbl = labels[row];
    const float* __restrict__ xr = x       + (size_t)row * FEAT_DIM;
    const float* __restrict__ cr = centers + (size_t)lbl * FEAT_DIM;

    // Warm the gathered center row toward L2 (global_prefetch_b8).
    __builtin_prefetch(cr, 0, 1);

    // Two independent WMMA accumulation chains to cover WMMA->WMMA latency;
    // matmul accumulation is linear, so summing the chains at the end is exact.
    v8f acc0 = {};
    v8f acc1 = {};

    // A-layout for 16x4 f32 (ISA 7.12.2): lane m (<16) holds row m, K=0,1;
    // lane 16+m holds row m, K=2,3. The matching 4x16 B layout for f*f^T is
    // the same per-lane data, so one fragment feeds both operands.
    #pragma unroll 4
    for (int kk = 0; kk < FEAT_DIM / 8; ++kk) {
        const int k0 = kk * 8 + half * 2;   // K-chunk 2*kk
        const int k1 = k0 + 4;              // K-chunk 2*kk + 1
        float2 xa = *(const float2*)(xr + k0);
        float2 ca = *(const float2*)(cr + k0);
        float2 xb = *(const float2*)(xr + k1);
        float2 cb = *(const float2*)(cr + k1);
        v2f fa; fa.x = xa.x - ca.x; fa.y = xa.y - ca.y;
        v2f fb; fb.x = xb.x - cb.x; fb.y = xb.y - cb.y;
        acc0 = __builtin_amdgcn_wmma_f32_16x16x4_f32(
            false, fa, false, fa, (short)0, acc0, false, false);
        acc1 = __builtin_amdgcn_wmma_f32_16x16x4_f32(
            false, fb, false, fb, (short)0, acc1, false, false);
    }

    // Diagonal extraction from the 16x16 f32 C/D layout:
    //   d_m (m<8)  lives in VGPR m,   lane m
    //   d_m (m>=8) lives in VGPR m-8, lane 24+(m-8)
    const int idx = (lane < 8) ? lane : ((lane >= 24) ? (lane - 24) : -1);
    float d = 0.0f;
    #pragma unroll
    for (int j = 0; j < 8; ++j) {
        float aj = acc0[j] + acc1[j];
        d = (idx == j) ? aj : d;
    }
    float contrib = (idx >= 0) ? fminf(fmaxf(d, CLAMP_MIN_F), CLAMP_MAX_F) : 0.0f;

    // wave32 butterfly reduction
    #pragma unroll
    for (int off = 16; off > 0; off >>= 1)
        contrib += __shfl_xor(contrib, off, 32);

    if (lane == 0) partials[blockIdx.x] = contrib;
}

__global__ void cmcl_finalize_kernel(const float* __restrict__ partials,
                                     float* __restrict__ out)
{
    // Fixed-order serial sum: bit-deterministic across graph replays.
    if (threadIdx.x == 0 && blockIdx.x == 0) {
        float s = 0.0f;
        for (int i = 0; i < NBLOCKS; ++i) s += partials[i];
        out[0] = s / (float)BATCH + (float)(NUM_CLASSES - 1) * CLAMP_MIN_F;
    }
}

extern "C" void kernel_launch(void* const* d_in, const int* in_sizes, int n_in,
                              void* d_out, int out_size, void* d_ws, size_t ws_size,
                              hipStream_t stream) {
    const float* x       = (const float*)d_in[0];
    const int*   labels  = (const int*)d_in[1];   // harness contract: integer -> const int*
    const float* centers = (const float*)d_in[2];
    float*       out     = (float*)d_out;
    float*       partials = (float*)d_ws;         // NBLOCKS floats of scratch

    cmcl_partial_kernel<<<NBLOCKS, 32, 0, stream>>>(x, labels, centers, partials);
    cmcl_finalize_kernel<<<1, 32, 0, stream>>>(partials, out);
}